// RoIpool_52132313038914
// MI455X (gfx1250) — compile-verified
//
#include <hip/hip_runtime.h>
#include <stdint.h>

// RoI-align (crop_and_resize 7x7 bilinear), fp32.
// feature_map: (16, 27, 48, 512), nmses: (16, 100, 4) -> out (16,100,7,7,512).
// Memory-bound: ~160MB NT writes + L2-resident gathers (FM = 42.5MB < 192MB L2).

#define POOLK 7
#define FH    27
#define FW    48
#define FC    512
#define NB    16
#define NR    100

typedef float f4  __attribute__((ext_vector_type(4)));
typedef int   v4i __attribute__((ext_vector_type(4)));

#if defined(__gfx1250__) && __has_builtin(__builtin_amdgcn_global_load_async_to_lds_b128)
#define USE_ASYNC 1
#else
#define USE_ASYNC 0
#endif

__global__ __launch_bounds__(128) void roi_align_kernel(
    const float* __restrict__ fm,
    const float* __restrict__ nms,
    float* __restrict__ out)
{
    const int pix = blockIdx.x;                 // (b*NR + r)*49 + py*7 + px
    const int t   = threadIdx.x;                // 0..127, owns 4 channels
    const int pp  = pix % (POOLK * POOLK);
    const int br  = pix / (POOLK * POOLK);      // b*NR + r
    const int py  = pp / POOLK;
    const int px  = pp % POOLK;
    const int b   = br / NR;

    // Box decode (matches reference): cx,cy,w,h -> normalized y1,x1,y2,x2
    const float cx = nms[br * 4 + 0];
    const float cy = nms[br * 4 + 1];
    const float bw = nms[br * 4 + 2];
    const float bh = nms[br * 4 + 3];

    const float x1 = (cx - 0.5f * bw) / 768.0f;
    const float x2 = (cx + 0.5f * bw) / 768.0f;
    const float y1 = (cy - 0.5f * bh) / 432.0f;
    const float y2 = (cy + 0.5f * bh) / 432.0f;

    const float ys = y1 * (float)(FH - 1) + (float)py * (y2 - y1) * (float)(FH - 1) / (float)(POOLK - 1);
    const float xs = x1 * (float)(FW - 1) + (float)px * (x2 - x1) * (float)(FW - 1) / (float)(POOLK - 1);

    const bool valid = (ys >= 0.0f) && (ys <= (float)(FH - 1)) &&
                       (xs >= 0.0f) && (xs <= (float)(FW - 1));

    const float y0f = floorf(ys);
    const float x0f = floorf(xs);
    const float ly  = ys - y0f;
    const float lx  = xs - x0f;

    int y0 = (int)y0f; y0 = y0 < 0 ? 0 : (y0 > FH - 1 ? FH - 1 : y0);
    int x0 = (int)x0f; x0 = x0 < 0 ? 0 : (x0 > FW - 1 ? FW - 1 : x0);
    const int y1i = (y0 + 1 > FH - 1) ? FH - 1 : y0 + 1;
    const int x1i = (x0 + 1 > FW - 1) ? FW - 1 : x0 + 1;

    const int    c    = t * 4;
    const size_t base = (size_t)b * (size_t)(FH * FW * FC);
    const float* ptl  = fm + base + ((size_t)y0  * FW + x0 ) * FC + c;
    const float* ptr_ = fm + base + ((size_t)y0  * FW + x1i) * FC + c;
    const float* pbl  = fm + base + ((size_t)y1i * FW + x0 ) * FC + c;
    const float* pbr  = fm + base + ((size_t)y1i * FW + x1i) * FC + c;

    f4 tl, tr, bl, brv;

#if USE_ASYNC
    // Stage the four 2KB corner rows into LDS with the CDNA5 async engine;
    // each lane consumes only its own slots, so a per-wave asynccnt wait suffices.
    __shared__ float lds[4 * FC];               // 8 KB
    typedef __attribute__((address_space(1))) v4i v4i_as1;   // global int4*
    typedef __attribute__((address_space(3))) v4i v4i_as3;   // LDS int4*
    __builtin_amdgcn_global_load_async_to_lds_b128(
        (v4i_as1*)(uintptr_t)ptl,  (v4i_as3*)(uintptr_t)&lds[0 * FC + c], 0, 0);
    __builtin_amdgcn_global_load_async_to_lds_b128(
        (v4i_as1*)(uintptr_t)ptr_, (v4i_as3*)(uintptr_t)&lds[1 * FC + c], 0, 0);
    __builtin_amdgcn_global_load_async_to_lds_b128(
        (v4i_as1*)(uintptr_t)pbl,  (v4i_as3*)(uintptr_t)&lds[2 * FC + c], 0, 0);
    __builtin_amdgcn_global_load_async_to_lds_b128(
        (v4i_as1*)(uintptr_t)pbr,  (v4i_as3*)(uintptr_t)&lds[3 * FC + c], 0, 0);
#if __has_builtin(__builtin_amdgcn_s_wait_asynccnt)
    __builtin_amdgcn_s_wait_asynccnt(0);
#else
    asm volatile("s_wait_asynccnt 0" ::: "memory");
#endif
    asm volatile("" ::: "memory");              // compiler ordering fence for LDS reads
    tl  = *(const f4*)&lds[0 * FC + c];
    tr  = *(const f4*)&lds[1 * FC + c];
    bl  = *(const f4*)&lds[2 * FC + c];
    brv = *(const f4*)&lds[3 * FC + c];
#else
    tl  = *(const f4*)ptl;
    tr  = *(const f4*)ptr_;
    bl  = *(const f4*)pbl;
    brv = *(const f4*)pbr;
#endif

    // Same lerp form as reference: a + (b - a) * t
    f4 top = tl + (tr  - tl)  * lx;
    f4 bot = bl + (brv - bl)  * lx;
    f4 o   = top + (bot - top) * ly;
    if (!valid) o = (f4)(0.0f);

    // Output is write-once streaming (160MB): NT store keeps FM resident in L2.
    float* outp = out + (size_t)pix * FC + c;
    __builtin_nontemporal_store(o, (f4*)outp);
}

extern "C" void kernel_launch(void* const* d_in, const int* in_sizes, int n_in,
                              void* d_out, int out_size, void* d_ws, size_t ws_size,
                              hipStream_t stream) {
    const float* fm  = (const float*)d_in[0];   // (16,27,48,512) fp32
    const float* nms = (const float*)d_in[1];   // (16,100,4)     fp32
    float* out = (float*)d_out;                 // (16,100,7,7,512) fp32
    (void)in_sizes; (void)n_in; (void)out_size; (void)d_ws; (void)ws_size;

    const int nblocks = NB * NR * POOLK * POOLK; // 78,400 pixels
    roi_align_kernel<<<dim3(nblocks), dim3(128), 0, stream>>>(fm, nms, out);
}